// DGNLayerComplex_86517821215490
// MI455X (gfx1250) — compile-verified
//
#include <hip/hip_runtime.h>
#include <math.h>
#include <stdint.h>

typedef __attribute__((ext_vector_type(2))) float v2f;
typedef __attribute__((ext_vector_type(8))) float v8f;

#define N_NODES 50000
#define N_EDGES 800000
#define IN_DIM 128
#define AVG_D_LOG 2.772588722239781f   // log(16)

// ---------- branchless float<->unsigned monotonic key (float order == unsigned key order) ----------
__device__ __forceinline__ unsigned int fkey(float v) {
    unsigned int u = __float_as_uint(v);
    return (u & 0x80000000u) ? ~u : (u | 0x80000000u);
}
__device__ __forceinline__ float funkey(unsigned int k) {
    unsigned int u = (k & 0x80000000u) ? (k ^ 0x80000000u) : ~k;
    return __uint_as_float(u);
}

// ---------- async global->LDS staging helper (CDNA5: tracked by ASYNCcnt) ----------
__device__ __forceinline__ void async_copy_b128(unsigned lds_addr, const void* gaddr) {
    unsigned long long ga = (unsigned long long)(size_t)gaddr;
    asm volatile("global_load_async_to_lds_b128 %0, %1, off"
                 :: "v"(lds_addr), "v"(ga) : "memory");
}
__device__ __forceinline__ void wait_asynccnt0() {
    asm volatile("s_wait_asynccnt 0x0" ::: "memory");
}

// ---------------- pack weights into K-paired layout (once per call, ~800KB) ----------------
// pre[k2*128+n]  = {Wpre[2k2][n],  Wpre[2k2+1][n]}   (136x128 v2f)
// post[k2*128+n] = {Wpost[2k2][n], Wpost[2k2+1][n]}  (640x128 v2f)
__global__ __launch_bounds__(256) void dgn_pack_kernel(const float* __restrict__ Wpre,
                                                       const float* __restrict__ Wpost,
                                                       v2f* __restrict__ wpre_p,
                                                       v2f* __restrict__ wpost_p) {
    int idx = blockIdx.x * 256 + threadIdx.x;
    if (idx < 136 * 128) {
        int k2 = idx >> 7, n = idx & 127;
        v2f w;
        w.x = Wpre[(2 * k2) * 128 + n];
        w.y = Wpre[(2 * k2 + 1) * 128 + n];
        wpre_p[idx] = w;
    }
    int idx2 = idx - 136 * 128;
    if (idx2 >= 0 && idx2 < 640 * 128) {
        int k2 = idx2 >> 7, n = idx2 & 127;
        v2f w;
        w.x = Wpost[(size_t)(2 * k2) * 128 + n];
        w.y = Wpost[(size_t)(2 * k2 + 1) * 128 + n];
        wpost_p[idx2] = w;
    }
}

// ---------------- init: sum=0, mx-key=0 (< key(-inf)), mn-key=~0 (> key(+inf)), deg=0 ----------------
__global__ __launch_bounds__(256) void dgn_init_kernel(float* __restrict__ sum,
                                                       unsigned int* __restrict__ mxk,
                                                       unsigned int* __restrict__ mnk,
                                                       int* __restrict__ deg) {
    int idx = blockIdx.x * 256 + threadIdx.x;
    if (idx < N_NODES * IN_DIM) {
        sum[idx] = 0.0f;
        mxk[idx] = 0u;
        mnk[idx] = 0xFFFFFFFFu;
    }
    if (idx < N_NODES) deg[idx] = 0;
}

// ---------------- edge pretrans GEMM (fp32 WMMA) + fused scatter reduce ----------------
// One wave computes a 16-edge x 128-out tile: D = A(16x272) * W_pre(272x128), A gathered from
// [h[src] | h[dst] | e]. K loop split into 3 branch-free phases. W_pre (K-paired, 139264 B)
// staged into LDS via global_load_async_to_lds_b128.
__global__ __launch_bounds__(256) void dgn_edge_kernel(
    const float* __restrict__ h, const float* __restrict__ ef,
    const int* __restrict__ src, const int* __restrict__ dst,
    const v2f* __restrict__ wpre_p, const float* __restrict__ bpre,
    float* __restrict__ sum, unsigned int* __restrict__ mxk,
    unsigned int* __restrict__ mnk, int* __restrict__ deg) {
    __shared__ v2f wp[136 * 128];   // 272x128 fp32 = 139264 B, paired along K

    const int tid = threadIdx.x;
    {
        unsigned lbase = (unsigned)(size_t)(&wp[0]);
        for (int i = tid; i < 8704; i += 256) {          // 8704 x 16 B = 139264 B
            async_copy_b128(lbase + (unsigned)i * 16u,
                            (const char*)wpre_p + (size_t)i * 16);
        }
        wait_asynccnt0();
    }
    __syncthreads();

    const int lane  = tid & 31;
    const int wave  = tid >> 5;
    const int lrow  = lane & 15;     // row (A) / col-within-tile (B,C)
    const int hi    = lane >> 4;     // 0: K=0,1 / M=r ; 1: K=2,3 / M=r+8
    const int khalf = hi * 2;

    for (int j = 0; j < 4; ++j) {
        int tile = blockIdx.x * 32 + j * 8 + wave;     // wave-uniform
        if (tile >= N_EDGES / 16) continue;
        int ebase = tile * 16;

        int edge = ebase + lrow;
        int sIdx = src[edge];
        int dIdx = dst[edge];
        const float* hs = h  + (size_t)sIdx * IN_DIM + khalf;
        const float* hd = h  + (size_t)dIdx * IN_DIM + khalf;
        const float* er = ef + (size_t)edge * 16 + khalf;

        int dnode[8];
#pragma unroll
        for (int r = 0; r < 8; ++r) dnode[r] = dst[ebase + hi * 8 + r];

        if (lane < 16) atomicAdd(&deg[dIdx], 1);   // one increment per edge

        v8f acc[8];
#pragma unroll
        for (int t = 0; t < 8; ++t) acc[t] = {};

        // ---- phase 1: k = 0..31, A from h[src] ----
#pragma unroll 4
        for (int k = 0; k < 32; ++k) {
            v2f a = *(const v2f*)(hs + 4 * k);
            int bbase = (2 * k + hi) * 128 + lrow;
#pragma unroll
            for (int t = 0; t < 8; ++t) {
                v2f b = wp[bbase + 16 * t];
                acc[t] = __builtin_amdgcn_wmma_f32_16x16x4_f32(
                    false, a, false, b, (short)0, acc[t], false, false);
            }
        }
        // ---- phase 2: k = 32..63, A from h[dst] ----
#pragma unroll 4
        for (int k = 32; k < 64; ++k) {
            v2f a = *(const v2f*)(hd + 4 * (k - 32));
            int bbase = (2 * k + hi) * 128 + lrow;
#pragma unroll
            for (int t = 0; t < 8; ++t) {
                v2f b = wp[bbase + 16 * t];
                acc[t] = __builtin_amdgcn_wmma_f32_16x16x4_f32(
                    false, a, false, b, (short)0, acc[t], false, false);
            }
        }
        // ---- phase 3: k = 64..67, A from edge features ----
#pragma unroll
        for (int k = 64; k < 68; ++k) {
            v2f a = *(const v2f*)(er + 4 * (k - 64));
            int bbase = (2 * k + hi) * 128 + lrow;
#pragma unroll
            for (int t = 0; t < 8; ++t) {
                v2f b = wp[bbase + 16 * t];
                acc[t] = __builtin_amdgcn_wmma_f32_16x16x4_f32(
                    false, a, false, b, (short)0, acc[t], false, false);
            }
        }

        // scatter: per edge, 128 outputs -> L2-resident atomics (add + branchless key max/min)
#pragma unroll
        for (int t = 0; t < 8; ++t) {
            int col = 16 * t + lrow;
            float bv = bpre[col];
#pragma unroll
            for (int r = 0; r < 8; ++r) {
                float val = acc[t][r] + bv;
                size_t off = (size_t)dnode[r] * IN_DIM + col;
                atomicAdd(&sum[off], val);
                unsigned int kk = fkey(val);
                atomicMax(&mxk[off], kk);
                atomicMin(&mnk[off], kk);
            }
        }
    }
}

// ---------------- finalize: mean in place, un-key + mask max/min, per-node scalers ----------------
__global__ __launch_bounds__(256) void dgn_finalize_kernel(
    float* __restrict__ sum, unsigned int* __restrict__ mxk,
    unsigned int* __restrict__ mnk, const int* __restrict__ deg,
    float* __restrict__ scal) {
    int idx = blockIdx.x * 256 + threadIdx.x;
    if (idx >= N_NODES * IN_DIM) return;
    int n = idx >> 7;
    int d = deg[n];
    float fd = (float)d;
    sum[idx] = sum[idx] / fmaxf(fd, 1.0f);
    bool has = d > 0;
    // convert keys back to floats in place (node kernel reads these buffers as float)
    mxk[idx] = __float_as_uint(has ? funkey(mxk[idx]) : 0.0f);
    mnk[idx] = __float_as_uint(has ? funkey(mnk[idx]) : 0.0f);
    if ((idx & 127) == 0) {
        float ld = logf(fd + 1.0f);
        scal[2 * n]     = ld / AVG_D_LOG;
        scal[2 * n + 1] = AVG_D_LOG / (ld > 0.0f ? ld : 1.0f);
    }
}

// ---------------- node posttrans GEMM (fp32 WMMA) + fused epilogue ----------------
// out[n] = [h | mean | mx | mn | a*(mean,mx,mn) | t*(mean,mx,mn)] @ W_post + b_post, then
// snorm, eval-BN, ReLU, residual. Ten K-blocks of 128; each 64 KB W_post block staged into
// LDS via global_load_async_to_lds_b128 from the pre-packed layout.
__global__ __launch_bounds__(256) void dgn_node_kernel(
    const float* __restrict__ h, const float* __restrict__ mean,
    const float* __restrict__ mx, const float* __restrict__ mn,
    const float* __restrict__ scal, const v2f* __restrict__ wpost_p,
    const float* __restrict__ bpost, const float* __restrict__ snorm,
    const float* __restrict__ gamma, const float* __restrict__ beta,
    float* __restrict__ out) {
    __shared__ v2f wl[64 * 128];    // one 128x128 K-block of W_post, paired along K (64 KB)

    const int tid   = threadIdx.x;
    const int lane  = tid & 31;
    const int wave  = tid >> 5;
    const int lrow  = lane & 15;
    const int hi    = lane >> 4;
    const int khalf = hi * 2;

    int tile   = blockIdx.x * 8 + wave;          // wave-uniform
    bool valid = tile < (N_NODES / 16);
    int nbase  = valid ? tile * 16 : 0;
    int nodeA  = nbase + lrow;
    float as = scal[2 * nodeA];
    float ts = scal[2 * nodeA + 1];

    v8f acc[8];
#pragma unroll
    for (int t = 0; t < 8; ++t) acc[t] = {};

    const unsigned lbase = (unsigned)(size_t)(&wl[0]);
    for (int b = 0; b < 10; ++b) {
        for (int i = tid; i < 4096; i += 256) {          // 4096 x 16 B = 64 KB
            async_copy_b128(lbase + (unsigned)i * 16u,
                            (const char*)wpost_p + (size_t)b * 65536 + (size_t)i * 16);
        }
        wait_asynccnt0();
        __syncthreads();

        if (valid) {
            const float* Aptr;
            float sc;
            switch (b) {
                case 0: Aptr = h;    sc = 1.0f; break;
                case 1: Aptr = mean; sc = 1.0f; break;
                case 2: Aptr = mx;   sc = 1.0f; break;
                case 3: Aptr = mn;   sc = 1.0f; break;
                case 4: Aptr = mean; sc = as;   break;
                case 5: Aptr = mx;   sc = as;   break;
                case 6: Aptr = mn;   sc = as;   break;
                case 7: Aptr = mean; sc = ts;   break;
                case 8: Aptr = mx;   sc = ts;   break;
                default: Aptr = mn;  sc = ts;   break;
            }
            const float* arow = Aptr + (size_t)nodeA * IN_DIM + khalf;
#pragma unroll 4
            for (int k = 0; k < 32; ++k) {
                v2f a = *(const v2f*)(arow + 4 * k);
                a = a * sc;
                int bbase = (2 * k + hi) * 128 + lrow;
#pragma unroll
                for (int t = 0; t < 8; ++t) {
                    v2f bb = wl[bbase + 16 * t];
                    acc[t] = __builtin_amdgcn_wmma_f32_16x16x4_f32(
                        false, a, false, bb, (short)0, acc[t], false, false);
                }
            }
        }
        __syncthreads();
    }

    if (!valid) return;
    const float bnscale = rsqrtf(1.0f + 1e-5f);
#pragma unroll
    for (int t = 0; t < 8; ++t) {
        int col = 16 * t + lrow;
        float bp = bpost[col];
        float g  = gamma[col] * bnscale;
        float bt = beta[col];
#pragma unroll
        for (int r = 0; r < 8; ++r) {
            int node = nbase + hi * 8 + r;
            float v = acc[t][r] + bp;
            v *= snorm[node];              // graph norm
            v = v * g + bt;                // eval-mode BN
            v = fmaxf(v, 0.0f);            // ReLU
            v += h[(size_t)node * IN_DIM + col];   // residual
            out[(size_t)node * IN_DIM + col] = v;
        }
    }
}

extern "C" void kernel_launch(void* const* d_in, const int* in_sizes, int n_in,
                              void* d_out, int out_size, void* d_ws, size_t ws_size,
                              hipStream_t stream) {
    const float* h     = (const float*)d_in[0];
    const float* e     = (const float*)d_in[1];
    const int*   src   = (const int*)d_in[2];
    const int*   dst   = (const int*)d_in[3];
    const float* snorm = (const float*)d_in[4];
    const float* Wpre  = (const float*)d_in[5];
    const float* bpre  = (const float*)d_in[6];
    const float* Wpost = (const float*)d_in[7];
    const float* bpost = (const float*)d_in[8];
    const float* gamma = (const float*)d_in[9];
    const float* beta  = (const float*)d_in[10];
    float* out = (float*)d_out;

    // workspace layout: sum | mxk | mnk | deg | scal | wpre_p | wpost_p  (~78.2 MB)
    float*        sum    = (float*)d_ws;
    unsigned int* mxk    = (unsigned int*)(sum + (size_t)N_NODES * IN_DIM);
    unsigned int* mnk    = mxk + (size_t)N_NODES * IN_DIM;
    int*          deg    = (int*)(mnk + (size_t)N_NODES * IN_DIM);
    float*        scal   = (float*)(deg + N_NODES);
    v2f*          wpre_p = (v2f*)(scal + (size_t)2 * N_NODES);
    v2f*          wpost_p = wpre_p + (size_t)136 * 128;

    // 0) pack weights into K-paired layout (contiguous for async LDS staging)
    dgn_pack_kernel<<<(136 * 128 + 640 * 128 + 255) / 256, 256, 0, stream>>>(
        Wpre, Wpost, wpre_p, wpost_p);
    // 1) init accumulators
    dgn_init_kernel<<<(N_NODES * IN_DIM + 255) / 256, 256, 0, stream>>>(sum, mxk, mnk, deg);
    // 2) edge GEMM + fused atomic scatter (50000 edge tiles, 32 per block)
    dgn_edge_kernel<<<(N_EDGES / 16 + 31) / 32, 256, 0, stream>>>(
        h, e, src, dst, wpre_p, bpre, sum, mxk, mnk, deg);
    // 3) finalize mean/max/min + degree scalers
    dgn_finalize_kernel<<<(N_NODES * IN_DIM + 255) / 256, 256, 0, stream>>>(
        sum, mxk, mnk, deg, scal);
    // 4) node GEMM + epilogue (3125 node tiles, 8 per block)
    dgn_node_kernel<<<(N_NODES / 16 + 7) / 8, 256, 0, stream>>>(
        h, sum, (const float*)mxk, (const float*)mnk, scal, wpost_p, bpost, snorm,
        gamma, beta, out);
}